// GAT_24773371363587
// MI455X (gfx1250) — compile-verified
//
#include <hip/hip_runtime.h>
#include <hip/hip_bf16.h>

typedef float v2f __attribute__((ext_vector_type(2)));
typedef float v8f __attribute__((ext_vector_type(8)));

#define NEG_SLOPE 0.2f

// ---------- float <-> ordered-int (monotone) for atomic float max ----------
__device__ __forceinline__ int f2ord(float f) {
    int i = __float_as_int(f);
    return (i >= 0) ? i : (i ^ 0x7fffffff);
}
__device__ __forceinline__ float ord2f(int i) {
    return __int_as_float((i >= 0) ? i : (i ^ 0x7fffffff));
}

// ---------------------------------------------------------------------------
// GEMM: Out[M x 128] = A[M x 128] * W[128 x 128] using V_WMMA_F32_16X16X4_F32.
// One wave computes one 16x16 tile; 4 waves/block share a 16x128 A stripe in
// LDS (row stride 132 to avoid bank conflicts on the per-K column gather).
// grid = (ceil(M/16), 2), block = 128 (4 waves -> 4 col tiles each of 16).
// ---------------------------------------------------------------------------
__global__ __launch_bounds__(128) void gat_gemm_wmma(
    const float* __restrict__ A, const float* __restrict__ W,
    float* __restrict__ Out, int M) {
    __shared__ float As[16][132];

    const int rowBase = blockIdx.x * 16;

    // Cooperative load of the 16x128 A stripe (clamp rows at the tail).
    #pragma unroll
    for (int i = 0; i < 16; ++i) {
        int idx = threadIdx.x + i * 128;   // 0..2047
        int r = idx >> 7;
        int k = idx & 127;
        int row = rowBase + r;
        if (row >= M) row = M - 1;
        As[r][k] = A[(size_t)row * 128 + k];
    }
    __syncthreads();

    const int wave = threadIdx.x >> 5;
    const int lane = threadIdx.x & 31;
    const int half = lane >> 4;   // 0: K pair {0,1}; 1: K pair {2,3}
    const int l16  = lane & 15;
    const int ncol = (blockIdx.y * 4 + wave) * 16 + l16;

    v8f acc = {};
    for (int kk = 0; kk < 128; kk += 4) {
        // A fragment (16x4): lane group selects K pair, l16 selects row M.
        v2f a, b;
        a.x = As[l16][kk + half * 2 + 0];
        a.y = As[l16][kk + half * 2 + 1];
        // B fragment (4x16): l16 selects column N, lane group selects K pair.
        b.x = W[(size_t)(kk + half * 2 + 0) * 128 + ncol];
        b.y = W[(size_t)(kk + half * 2 + 1) * 128 + ncol];
        acc = __builtin_amdgcn_wmma_f32_16x16x4_f32(
            /*neg_a=*/false, a, /*neg_b=*/false, b,
            /*c_mod=*/(short)0, acc, /*reuse_a=*/false, /*reuse_b=*/false);
    }

    // D layout: VGPR r -> row r (lanes 0-15) / row r+8 (lanes 16-31), col = l16.
    #pragma unroll
    for (int r = 0; r < 8; ++r) {
        int row = rowBase + r + half * 8;
        if (row < M) Out[(size_t)row * 128 + ncol] = acc[r];
    }
}

// ---------------------------------------------------------------------------
// Attention coefficients: el[n,h] = sum_d feat[n,h,d]*al[h,d] (same for er).
// feat is [N, 2*64] row-major, al/ar are [2*64].
// ---------------------------------------------------------------------------
__global__ void gat_coef(const float* __restrict__ feat,
                         const float* __restrict__ al,
                         const float* __restrict__ ar,
                         float* __restrict__ el, float* __restrict__ er,
                         int N) {
    int n = blockIdx.x * blockDim.x + threadIdx.x;
    if (n >= N) return;
    const float* f = feat + (size_t)n * 128;
    float e0 = 0.f, e1 = 0.f, r0 = 0.f, r1 = 0.f;
    #pragma unroll 4
    for (int d = 0; d < 64; ++d) {
        float h0 = f[d], h1 = f[64 + d];
        e0 += h0 * al[d];
        e1 += h1 * al[64 + d];
        r0 += h0 * ar[d];
        r1 += h1 * ar[64 + d];
    }
    el[n * 2 + 0] = e0; el[n * 2 + 1] = e1;
    er[n * 2 + 0] = r0; er[n * 2 + 1] = r1;
}

// ---------------------------------------------------------------------------
// Per-layer init: mkey = ordered(-inf), s = 0 for N*2 entries.
// ---------------------------------------------------------------------------
__global__ void gat_init_ms(int* __restrict__ mkey, float* __restrict__ s, int n2) {
    int i = blockIdx.x * blockDim.x + threadIdx.x;
    if (i < n2) { mkey[i] = 0x80000000; s[i] = 0.f; }
}

// out[n, f] = bias[f] for f in [0,128)
__global__ void gat_init_bias(float* __restrict__ out, const float* __restrict__ b,
                              int total) {
    int i = blockIdx.x * blockDim.x + threadIdx.x;
    if (i < total) out[i] = b[i & 127];
}

__device__ __forceinline__ void edge_sd(long long i, long long E,
                                        const int* __restrict__ src,
                                        const int* __restrict__ dst,
                                        int& s, int& d) {
    if (i < E) { s = src[i]; d = dst[i]; }
    else       { s = (int)(i - E); d = s; }   // self loop
}

// segment max of leaky(e) over dst, via ordered-int atomicMax
__global__ void gat_edge_max(const int* __restrict__ src, const int* __restrict__ dst,
                             long long E, long long Et,
                             const float* __restrict__ el, const float* __restrict__ er,
                             int* __restrict__ mkey) {
    long long i = (long long)blockIdx.x * blockDim.x + threadIdx.x;
    if (i >= Et) return;
    int s, d; edge_sd(i, E, src, dst, s, d);
    #pragma unroll
    for (int h = 0; h < 2; ++h) {
        float e = el[s * 2 + h] + er[d * 2 + h];
        e = (e >= 0.f) ? e : NEG_SLOPE * e;
        atomicMax(&mkey[d * 2 + h], f2ord(e));
    }
}

// segment sum of exp(e - m[dst])
__global__ void gat_edge_sum(const int* __restrict__ src, const int* __restrict__ dst,
                             long long E, long long Et,
                             const float* __restrict__ el, const float* __restrict__ er,
                             const int* __restrict__ mkey, float* __restrict__ ssum) {
    long long i = (long long)blockIdx.x * blockDim.x + threadIdx.x;
    if (i >= Et) return;
    int s, d; edge_sd(i, E, src, dst, s, d);
    #pragma unroll
    for (int h = 0; h < 2; ++h) {
        float e = el[s * 2 + h] + er[d * 2 + h];
        e = (e >= 0.f) ? e : NEG_SLOPE * e;
        atomicAdd(&ssum[d * 2 + h], expf(e - ord2f(mkey[d * 2 + h])));
    }
}

// aggregation: one wave per edge; lane handles 4 contiguous features.
__global__ __launch_bounds__(256) void gat_edge_agg(
    const int* __restrict__ src, const int* __restrict__ dst,
    long long E, long long Et,
    const float* __restrict__ el, const float* __restrict__ er,
    const int* __restrict__ mkey, const float* __restrict__ ssum,
    const float* __restrict__ feat, float* __restrict__ out) {
    long long w = (long long)blockIdx.x * 8 + (threadIdx.x >> 5);
    if (w >= Et) return;
    const int lane = threadIdx.x & 31;
    int s, d; edge_sd(w, E, src, dst, s, d);

    const int f0 = lane * 4;        // 0..124
    const int h  = f0 >> 6;         // head 0 or 1
    float e = el[s * 2 + h] + er[d * 2 + h];
    e = (e >= 0.f) ? e : NEG_SLOPE * e;
    const float alpha = expf(e - ord2f(mkey[d * 2 + h])) / ssum[d * 2 + h];

    const float4 v = *(const float4*)(feat + (size_t)s * 128 + f0);
    float* o = out + (size_t)d * 128 + f0;
    atomicAdd(o + 0, alpha * v.x);
    atomicAdd(o + 1, alpha * v.y);
    atomicAdd(o + 2, alpha * v.z);
    atomicAdd(o + 3, alpha * v.w);
}

__global__ void gat_relu(float* __restrict__ x, int total) {
    int i = blockIdx.x * blockDim.x + threadIdx.x;
    if (i < total) x[i] = fmaxf(x[i], 0.f);
}

// ---------------------------------------------------------------------------
extern "C" void kernel_launch(void* const* d_in, const int* in_sizes, int n_in,
                              void* d_out, int out_size, void* d_ws, size_t ws_size,
                              hipStream_t stream) {
    const float* x   = (const float*)d_in[0];
    const int*   src = (const int*)  d_in[1];
    const int*   dst = (const int*)  d_in[2];
    const float* W1  = (const float*)d_in[3];
    const float* al1 = (const float*)d_in[4];
    const float* ar1 = (const float*)d_in[5];
    const float* b1  = (const float*)d_in[6];
    const float* W2  = (const float*)d_in[7];
    const float* al2 = (const float*)d_in[8];
    const float* ar2 = (const float*)d_in[9];
    const float* b2  = (const float*)d_in[10];

    const int N = in_sizes[0] / 128;
    const long long E  = in_sizes[1];
    const long long Et = E + N;          // with self loops

    float* out = (float*)d_out;

    // workspace layout
    float* feat = (float*)d_ws;          // [N*128] pre-aggregation features
    float* el   = feat + (size_t)N * 128;
    float* er   = el + (size_t)N * 2;
    float* ssum = er + (size_t)N * 2;
    int*   mkey = (int*)(ssum + (size_t)N * 2);

    const dim3 gemmGrid((N + 15) / 16, 2);
    const int  totalNF = N * 128;
    const int  n2      = N * 2;
    const int  eBlocks = (int)((Et + 255) / 256);
    const int  aBlocks = (int)((Et + 7) / 8);    // 8 waves (edges) per 256-thread block

    // ---------------- Layer 1 ----------------
    gat_gemm_wmma<<<gemmGrid, 128, 0, stream>>>(x, W1, feat, N);
    gat_coef<<<(N + 255) / 256, 256, 0, stream>>>(feat, al1, ar1, el, er, N);
    gat_init_ms<<<(n2 + 255) / 256, 256, 0, stream>>>(mkey, ssum, n2);
    gat_init_bias<<<(totalNF + 255) / 256, 256, 0, stream>>>(out, b1, totalNF);
    gat_edge_max<<<eBlocks, 256, 0, stream>>>(src, dst, E, Et, el, er, mkey);
    gat_edge_sum<<<eBlocks, 256, 0, stream>>>(src, dst, E, Et, el, er, mkey, ssum);
    gat_edge_agg<<<aBlocks, 256, 0, stream>>>(src, dst, E, Et, el, er, mkey, ssum, feat, out);
    gat_relu<<<(totalNF + 255) / 256, 256, 0, stream>>>(out, totalNF);

    // ---------------- Layer 2 ----------------
    gat_gemm_wmma<<<gemmGrid, 128, 0, stream>>>(out, W2, feat, N);   // consumes ReLU'd h
    gat_coef<<<(N + 255) / 256, 256, 0, stream>>>(feat, al2, ar2, el, er, N);
    gat_init_ms<<<(n2 + 255) / 256, 256, 0, stream>>>(mkey, ssum, n2);
    gat_init_bias<<<(totalNF + 255) / 256, 256, 0, stream>>>(out, b2, totalNF);
    gat_edge_max<<<eBlocks, 256, 0, stream>>>(src, dst, E, Et, el, er, mkey);
    gat_edge_sum<<<eBlocks, 256, 0, stream>>>(src, dst, E, Et, el, er, mkey, ssum);
    gat_edge_agg<<<aBlocks, 256, 0, stream>>>(src, dst, E, Et, el, er, mkey, ssum, feat, out);
}